// cluster_layer_3942779978386
// MI455X (gfx1250) — compile-verified
//
#include <hip/hip_runtime.h>

typedef __attribute__((ext_vector_type(2))) float v2f;
typedef __attribute__((ext_vector_type(8))) float v8f;

#define N_PTS      1000000
#define DIMS       64
#define KCL        20
#define LDS_STRIDE 68   // 64 + 4 pad: per-lane b64 B-reads hit 64 distinct banks

// Fast reciprocal: single v_rcp_f32 (TRANS op, co-issues with VALU) instead of
// hipcc's ~8-instruction IEEE-precise divide expansion. ~1 ulp is plenty for
// the soft-assignment nonlinearity.
__device__ __forceinline__ float fast_rcp(float x) {
  return __builtin_amdgcn_rcpf(x);
}

// Each wave: 16 rows of X vs 32 (padded) clusters via V_WMMA_F32_16X16X4_F32.
// Block = 128 threads = 4 waves = 64 rows. Grid = 1e6/64 = 15625 (exact).
__global__ __launch_bounds__(128) void cluster_q_wmma_kernel(
    const float* __restrict__ X, const float* __restrict__ C,
    float* __restrict__ out) {
  __shared__ float lds[32 * LDS_STRIDE + 32];
  float* csq = lds + 32 * LDS_STRIDE;
  const int tid = threadIdx.x;

  // Zero all of LDS (covers pad columns and phantom cluster rows 20..31).
  for (int i = tid; i < 32 * LDS_STRIDE + 32; i += 128) lds[i] = 0.0f;
  __syncthreads();
  // Stage clusters (20x64) into padded LDS, coalesced.
  for (int i = tid; i < KCL * DIMS; i += 128) {
    int r = i >> 6, c = i & 63;
    lds[r * LDS_STRIDE + c] = C[i];
  }
  __syncthreads();
  // ||c_k||^2 for k=0..31 (phantom rows give 0).
  if (tid < 32) {
    float s = 0.0f;
#pragma unroll
    for (int d = 0; d < DIMS; ++d) {
      float v = lds[tid * LDS_STRIDE + d];
      s += v * v;
    }
    csq[tid] = s;
  }
  __syncthreads();

  const int wave = tid >> 5;
  const int lane = tid & 31;
  const int lo   = lane & 15;  // row (A) / column (B,C)
  const int hi   = lane >> 4;  // K-half select for A/B operands
  const int rowBase = blockIdx.x * 64 + wave * 16;

  // A tile: 16 rows x 64 cols of X in WMMA 16x4 f32 layout:
  // lane(lo,hi) holds X[rowBase+lo][4c + 2*hi + {0,1}] for chunk c.
  const float* xrow = X + (size_t)(rowBase + lo) * DIMS + 2 * hi;
  v2f a[16];
  float xs = 0.0f;
#pragma unroll
  for (int c = 0; c < 16; ++c) {
    a[c] = *(const v2f*)(xrow + 4 * c);
    xs += a[c].x * a[c].x + a[c].y * a[c].y;
  }
  xs += __shfl_xor(xs, 16, 32);  // lanes L, L^16 hold the two K-halves of row lo

  // Cross terms: D=64 as 16 chunks of K=4; two 16-wide column tiles.
  v8f acc0 = {};
  v8f acc1 = {};
  const float* b0p = lds + lo * LDS_STRIDE + 2 * hi;         // clusters 0..15
  const float* b1p = lds + (16 + lo) * LDS_STRIDE + 2 * hi;  // clusters 16..31
#pragma unroll
  for (int c = 0; c < 16; ++c) {
    v2f b0 = *(const v2f*)(b0p + 4 * c);
    v2f b1 = *(const v2f*)(b1p + 4 * c);
    acc0 = __builtin_amdgcn_wmma_f32_16x16x4_f32(false, a[c], false, b0,
                                                 (short)0, acc0, false, false);
    acc1 = __builtin_amdgcn_wmma_f32_16x16x4_f32(false, a[c], false, b1,
                                                 (short)0, acc1, false, false);
  }

  // Epilogue: C/D layout => VGPR r holds row (r + 8*hi), lane lo = column.
  const float csq0 = csq[lo];
  const float csq1 = csq[16 + lo];
#pragma unroll
  for (int r = 0; r < 8; ++r) {
    const int m = r + 8 * hi;
    float xsm = __shfl(xs, m, 32);  // ||x_m||^2 (lane m holds it)
    float d0 = xsm + csq0 - 2.0f * acc0[r];
    float d1 = xsm + csq1 - 2.0f * acc1[r];
    float q0 = fast_rcp(1.0f + d0);
    float q1 = (lo < 4) ? fast_rcp(1.0f + d1) : 0.0f;  // k>=20 contributes 0
    float s = q0 + q1;
    s += __shfl_xor(s, 1, 32);
    s += __shfl_xor(s, 2, 32);
    s += __shfl_xor(s, 4, 32);
    s += __shfl_xor(s, 8, 32);  // row sum over the 16-lane half
    float inv = fast_rcp(s);
    float* orow = out + (size_t)(rowBase + m) * KCL;
    orow[lo] = q0 * inv;
    if (lo < 4) orow[16 + lo] = q1 * inv;
  }
}

extern "C" void kernel_launch(void* const* d_in, const int* in_sizes, int n_in,
                              void* d_out, int out_size, void* d_ws,
                              size_t ws_size, hipStream_t stream) {
  const float* X = (const float*)d_in[0];   // (1e6, 64) f32
  const float* C = (const float*)d_in[1];   // (20, 64)  f32
  float* out = (float*)d_out;               // (1e6, 20) f32
  (void)in_sizes; (void)n_in; (void)out_size; (void)d_ws; (void)ws_size;
  cluster_q_wmma_kernel<<<dim3(N_PTS / 64), dim3(128), 0, stream>>>(X, C, out);
}